// LinearEncoder_21869973471803
// MI455X (gfx1250) — compile-verified
//
#include <hip/hip_runtime.h>

typedef __attribute__((ext_vector_type(2))) float v2f;
typedef __attribute__((ext_vector_type(4))) float v4f;
typedef __attribute__((ext_vector_type(8))) float v8f;

#define B_    128
#define HW_   196
#define HID_  768
#define K_    768          // 3 * 16 * 16
#define P_    16
#define IMG_  224
#define NPX_  14           // patches per image row/col

#define BM 128             // M tile per workgroup
#define BN 128             // N tile per workgroup
#define KT 16              // K chunk staged in LDS

// GEMM view of the patch-embed conv + per-sample patch shuffle:
//   out[m, n] = sum_k A[m, k] * W[n, k] + bias[n]
// where row m = i*196 + j gathers patch perm[m] of sample i = m/196.
__global__ __launch_bounds__(256) void patch_embed_gemm(
    const float* __restrict__ x,     // [128, 3, 224, 224]
    const float* __restrict__ W,     // [768, 3, 16, 16] == [768, 768] row-major
    const float* __restrict__ bias,  // [768]
    const int*   __restrict__ perm,  // [128, 196] flat
    float* __restrict__ out)         // [128, 196, 768]
{
    __shared__ float As[BM * KT];        // A tile, [m][k], row stride KT
    // B tile, k-pair interleaved: element (k, n) at [(k/2)*2*BN + 2*n + (k&1)]
    // -> the two K-values a lane needs for one WMMA B-frag are adjacent 8B.
    __shared__ float Bs[KT * BN];

    const int tid  = threadIdx.x;
    const int lane = tid & 31;
    const int wid  = tid >> 5;       // 0..7
    const int wm   = wid & 3;        // 4 waves along M  -> 32 rows each
    const int wn   = wid >> 2;       // 2 waves along N  -> 64 cols each

    const int n0 = blockIdx.x * BN;  // 6 N-blocks (fast-varying -> L2 reuse of A)
    const int m0 = blockIdx.y * BM;  // 196 M-blocks

    // ---- A-tile loader setup: 2 threads per row, 8 floats each ----
    const int arow  = tid >> 1;          // 0..127  local row
    const int ahalf = (tid & 1) * 8;     // dx offset 0 or 8
    const int m     = m0 + arow;
    const int samp  = m / HW_;
    const int pp    = perm[m];           // gathered patch id in [0,196)
    const int ph    = pp / NPX_;
    const int pw    = pp - ph * NPX_;
    const float* xpatch = x + (size_t)samp * (3 * IMG_ * IMG_)
                            + (size_t)(ph * P_) * IMG_ + (size_t)(pw * P_);

    // ---- B-tile loader setup: 2 threads per W row, 8 floats each ----
    const int brow  = tid >> 1;          // local n, 0..127
    const int bhalf = (tid & 1) * 8;     // k offset 0 or 8
    const float* wrow = W + (size_t)(n0 + brow) * K_;

    v8f acc[2][4];
    #pragma unroll
    for (int mi = 0; mi < 2; ++mi)
        #pragma unroll
        for (int ni = 0; ni < 4; ++ni)
            acc[mi][ni] = (v8f){0.f, 0.f, 0.f, 0.f, 0.f, 0.f, 0.f, 0.f};

    const int klo  = (lane >> 4) << 1;   // K sub-offset per lane half (0 or 2)
    const int lm   = lane & 15;          // M (for A frag) / N (for B frag) lane index

    for (int k0 = 0; k0 < K_; k0 += KT) {
        // ---- stage A chunk: k0 -> (channel c, patch row rr), 16 contiguous floats/row
        {
            const int c  = k0 >> 8;            // k0 / 256
            const int rr = (k0 & 255) >> 4;    // (k0 % 256) / 16
            const float* src = xpatch + (size_t)c * (IMG_ * IMG_)
                                      + (size_t)rr * IMG_ + ahalf;
            v4f a0 = *(const v4f*)(src);
            v4f a1 = *(const v4f*)(src + 4);
            *(v4f*)&As[arow * KT + ahalf]     = a0;
            *(v4f*)&As[arow * KT + ahalf + 4] = a1;
        }
        // ---- stage B chunk, transposed into k-pair-interleaved [k/2][n][2]
        {
            v4f w0 = *(const v4f*)(wrow + k0 + bhalf);
            v4f w1 = *(const v4f*)(wrow + k0 + bhalf + 4);
            const int kp = bhalf >> 1;         // k-pair index: 0 or 4
            v2f p;
            p.x = w0[0]; p.y = w0[1];
            *(v2f*)&Bs[(kp + 0) * (2 * BN) + 2 * brow] = p;
            p.x = w0[2]; p.y = w0[3];
            *(v2f*)&Bs[(kp + 1) * (2 * BN) + 2 * brow] = p;
            p.x = w1[0]; p.y = w1[1];
            *(v2f*)&Bs[(kp + 2) * (2 * BN) + 2 * brow] = p;
            p.x = w1[2]; p.y = w1[3];
            *(v2f*)&Bs[(kp + 3) * (2 * BN) + 2 * brow] = p;
        }
        // ---- prefetch next chunk while we compute this one
        if (k0 + KT < K_) {
            const int k1  = k0 + KT;
            const int c1  = k1 >> 8;
            const int rr1 = (k1 & 255) >> 4;
            __builtin_prefetch(xpatch + (size_t)c1 * (IMG_ * IMG_)
                                      + (size_t)rr1 * IMG_ + ahalf, 0, 3);
            __builtin_prefetch(wrow + k1 + bhalf, 0, 3);
        }
        __syncthreads();

        // ---- 4 WMMA K-steps of 4 over the staged chunk
        #pragma unroll
        for (int kk = 0; kk < KT; kk += 4) {
            // A frags: lane l -> M = l%16, VGPR v holds K = kk + (l/16)*2 + v
            v2f afrag[2];
            #pragma unroll
            for (int mi = 0; mi < 2; ++mi) {
                const int mrow = wm * 32 + mi * 16 + lm;
                afrag[mi] = *(const v2f*)&As[mrow * KT + kk + klo];
            }
            // B frags: lane l -> N = l%16, VGPR v holds K = kk + (l/16)*2 + v
            // k-pair (kk+klo, kk+klo+1) is contiguous in Bs -> one b64 load.
            const int krow = (kk + klo) >> 1;  // even pair index
            #pragma unroll
            for (int ni = 0; ni < 4; ++ni) {
                const int ncol = wn * 64 + ni * 16 + lm;
                v2f bfrag = *(const v2f*)&Bs[krow * (2 * BN) + 2 * ncol];
                #pragma unroll
                for (int mi = 0; mi < 2; ++mi)
                    acc[mi][ni] = __builtin_amdgcn_wmma_f32_16x16x4_f32(
                        false, afrag[mi], false, bfrag,
                        (short)0, acc[mi][ni], false, false);
            }
        }
        __syncthreads();
    }

    // ---- epilogue: bias add + store (C layout: VGPR r -> M = r + (l/16)*8, N = l%16)
    #pragma unroll
    for (int ni = 0; ni < 4; ++ni) {
        const int n  = n0 + wn * 64 + ni * 16 + lm;
        const float bv = bias[n];
        #pragma unroll
        for (int mi = 0; mi < 2; ++mi) {
            #pragma unroll
            for (int r = 0; r < 8; ++r) {
                const int mrow = m0 + wm * 32 + mi * 16 + ((lane >> 4) << 3) + r;
                out[(size_t)mrow * HID_ + n] = acc[mi][ni][r] + bv;
            }
        }
    }
}

extern "C" void kernel_launch(void* const* d_in, const int* in_sizes, int n_in,
                              void* d_out, int out_size, void* d_ws, size_t ws_size,
                              hipStream_t stream) {
    const float* x    = (const float*)d_in[0];
    const float* W    = (const float*)d_in[1];
    const float* bias = (const float*)d_in[2];
    const int*   perm = (const int*)d_in[3];
    float*       out  = (float*)d_out;

    dim3 grid(HID_ / BN, (B_ * HW_) / BM);   // (6, 196)
    patch_embed_gemm<<<grid, dim3(256), 0, stream>>>(x, W, bias, perm, out);
}